// QNNModel_55250459296299
// MI455X (gfx1250) — compile-verified
//
#include <hip/hip_runtime.h>
#include <math.h>

typedef float v2f __attribute__((ext_vector_type(2)));
typedef float v8f __attribute__((ext_vector_type(8)));

#define NQ      12
#define DIM     4096      // 2^12 amplitudes
#define NLAYERS 4
#define PPL     36        // per layer: 12 RY + 12 XX + 12 YY
#define THREADS 256
#define NPAIR   2048
#define NANG    (NQ + NLAYERS * PPL)   // 156 half-angle trig pairs per block

// Apply (RY(a0) (x) RY(a1) (x) RY(a2) (x) RY(a3)) on the 4-qubit group g
// (group g covers qubits 4g..4g+3, i.e. amplitude-index bits [11-4g .. 8-4g]).
// Real 16x16 matrix M applied in-place to re[] and im[] via chained
// V_WMMA_F32_16X16X4_F32: state viewed as 16 x 256, tiled into 16 column
// tiles of 16; each wave owns 4 (component,tile) jobs -> fully disjoint,
// no intra-stage races. Caller barriers before/after.
// cs/sn point at 4 precomputed cos(a/2)/sin(a/2) values in LDS.
__device__ __forceinline__ void apply_group_ry(float* re, float* im, int g,
                                               const float* cs, const float* sn,
                                               int lane, int wave)
{
    float c4[4], s4[4];
#pragma unroll
    for (int j = 0; j < 4; ++j) { c4[j] = cs[j]; s4[j] = sn[j]; }

    const int m    = lane & 15;
    const int half = lane >> 4;       // 0: lanes 0-15, 1: lanes 16-31

    // Build A chunks: A_k = M[:, 4k..4k+3] in the 16x4 f32 WMMA layout:
    // lane m holds rows; VGPR0 (.x) = K = 2*half, VGPR1 (.y) = K = 2*half+1.
    // M[t][t'] = prod_j f(t_b, t'_b) with f: (0,0)->c, (0,1)->-s, (1,0)->s, (1,1)->c
    // (nibble bit (3-j) belongs to qubit 4g+j since qubit q sits at index bit 11-q).
    v2f A[4];
#pragma unroll
    for (int k = 0; k < 4; ++k) {
        const int tp0 = 4 * k + 2 * half;
        const int tp1 = tp0 + 1;
        float v0 = 1.f, v1 = 1.f;
#pragma unroll
        for (int j = 0; j < 4; ++j) {
            const int mb  = (m   >> (3 - j)) & 1;
            const int t0b = (tp0 >> (3 - j)) & 1;
            const int t1b = (tp1 >> (3 - j)) & 1;
            v0 *= (mb == t0b) ? c4[j] : (mb ? s4[j] : -s4[j]);
            v1 *= (mb == t1b) ? c4[j] : (mb ? s4[j] : -s4[j]);
        }
        A[k].x = v0;
        A[k].y = v1;
    }

    const int sh     = 8 - 4 * g;         // shift of the group nibble in the index
    const int stride = 1 << sh;

    // 32 jobs: comp (re/im) x 16 column tiles; 8 waves -> 4 jobs each.
    for (int job = wave; job < 32; job += 8) {
        float* X = (job & 16) ? im : re;
        const int ct   = job & 15;
        const int n    = (lane & 15) + (ct << 4);                  // column 0..255
        const int base = ((n >> sh) << (sh + 4)) | (n & (stride - 1));

        v8f acc = {0.f, 0.f, 0.f, 0.f, 0.f, 0.f, 0.f, 0.f};
#pragma unroll
        for (int k = 0; k < 4; ++k) {
            const int r = 4 * k + 2 * half;    // B 4x16 layout mirrors A
            v2f Bv;
            Bv.x = X[base + r * stride];
            Bv.y = X[base + (r + 1) * stride];
            acc = __builtin_amdgcn_wmma_f32_16x16x4_f32(
                /*neg_a=*/false, A[k], /*neg_b=*/false, Bv,
                /*c_mod=*/(short)0, acc, /*reuse_a=*/false, /*reuse_b=*/false);
        }
        // D layout: VGPR v -> row v + 8*half, col = lane&15 (same columns we read)
#pragma unroll
        for (int v = 0; v < 8; ++v) {
            X[base + (v + 8 * half) * stride] = acc[v];
        }
    }
}

// exp(-i*theta/2 * P(x)P) with P in {X,Y} on ring neighbors (q, (q+1)%12).
// Each amplitude couples only with its double-bit-flip partner; canonical
// index has the high mask bit = 0, and in the YY case both pair members use
// the same +/- sin sign, so one thread updates the whole pair.
__device__ __forceinline__ void coupler(float* re, float* im, int q,
                                        float c, float s, bool yy, int tid)
{
    int hi, lo;
    if (q < 11) { hi = 11 - q; lo = 10 - q; }   // qubit q -> bit 11-q (hi), q+1 -> bit 10-q
    else        { hi = 11;     lo = 0;      }   // pair (11, 0): qubit0 -> bit 11, qubit11 -> bit 0
    const int mask = (1 << hi) | (1 << lo);
    const int lowm = (1 << hi) - 1;

#pragma unroll
    for (int p = tid; p < NPAIR; p += THREADS) {
        const int i = ((p & ~lowm) << 1) | (p & lowm);   // insert 0 at bit hi
        const int j = i ^ mask;
        // YY sign: -1 when the two bits of the index are equal; for canonical i
        // (hi bit 0) this is -1 iff lo bit is 0, and identical for both i and j.
        const float se = yy ? (((i >> lo) & 1) ? s : -s) : s;

        const float ri = re[i], mi = im[i];
        const float rj = re[j], mj = im[j];
        // new = c*amp - i*se*partner  =>  re += se*Im(partner), im -= se*Re(partner)
        re[i] = c * ri + se * mj;
        im[i] = c * mi - se * rj;
        re[j] = c * rj + se * mi;
        im[j] = c * mj - se * ri;
    }
}

__launch_bounds__(THREADS)
__global__ void qsim_kernel(const float* __restrict__ x_batch,
                            const float* __restrict__ qparams,
                            float* __restrict__ out)
{
    __shared__ float re[DIM];
    __shared__ float im[DIM];
    __shared__ float csh[NANG];   // cos(angle/2): [0..11]=embed, [12+l*36+k]=layer l
    __shared__ float snh[NANG];   // sin(angle/2)
    __shared__ float red[THREADS];

    const int tid  = threadIdx.x;
    const int b    = blockIdx.x;
    const int lane = tid & 31;
    const int wave = tid >> 5;

    // ---- init: |0...0>, plus ALL trig at a single site (one sinf/cosf
    // instantiation for the whole kernel; angles are block-uniform constants).
    for (int i = tid; i < DIM; i += THREADS) { re[i] = 0.f; im[i] = 0.f; }
    if (tid == 0) re[0] = 1.f;
    if (tid < NANG) {
        const float a = (tid < NQ) ? x_batch[b * NQ + tid] : qparams[tid - NQ];
        csh[tid] = cosf(0.5f * a);
        snh[tid] = sinf(0.5f * a);
    }
    __syncthreads();

    // ---- embedding: RY(x[b,q]) on every qubit, fused per 4-qubit group ----
#pragma unroll
    for (int g = 0; g < 3; ++g) {
        apply_group_ry(re, im, g, &csh[4 * g], &snh[4 * g], lane, wave);
        __syncthreads();
    }

    // ---- layers ----
    for (int l = 0; l < NLAYERS; ++l) {
        const int pb = NQ + l * PPL;
#pragma unroll
        for (int g = 0; g < 3; ++g) {
            apply_group_ry(re, im, g, &csh[pb + 4 * g], &snh[pb + 4 * g], lane, wave);
            __syncthreads();
        }
        for (int q = 0; q < NQ; ++q) {          // XX ring
            coupler(re, im, q, csh[pb + 12 + q], snh[pb + 12 + q], false, tid);
            __syncthreads();
        }
        for (int q = 0; q < NQ; ++q) {          // YY ring
            coupler(re, im, q, csh[pb + 24 + q], snh[pb + 24 + q], true, tid);
            __syncthreads();
        }
    }

    // ---- readout: <Z on qubit 0> = sum(bit11==0 |a|^2) - sum(bit11==1 |a|^2) ----
    float part = 0.f;
    const int i0 = tid * 16;
#pragma unroll
    for (int i = 0; i < 16; ++i) {
        const float r = re[i0 + i], q_ = im[i0 + i];
        part += r * r + q_ * q_;
    }
    if (tid >= 128) part = -part;               // indices >= 2048 have bit11 = 1
    red[tid] = part;
    __syncthreads();
    for (int off = 128; off > 0; off >>= 1) {
        if (tid < off) red[tid] += red[tid + off];
        __syncthreads();
    }
    if (tid == 0) out[b] = red[0];
}

extern "C" void kernel_launch(void* const* d_in, const int* in_sizes, int n_in,
                              void* d_out, int out_size, void* d_ws, size_t ws_size,
                              hipStream_t stream) {
    const float* x_batch = (const float*)d_in[0];   // [B, 12] fp32
    const float* qparams = (const float*)d_in[1];   // [144] fp32
    float* out = (float*)d_out;                     // [B] fp32
    const int batch = in_sizes[0] / NQ;             // 512
    qsim_kernel<<<batch, THREADS, 0, stream>>>(x_batch, qparams, out);
}